// Encoder2_46763603919351
// MI455X (gfx1250) — compile-verified
//
#include <hip/hip_runtime.h>

typedef float v2f __attribute__((ext_vector_type(2)));
typedef float v4f __attribute__((ext_vector_type(4)));
typedef float v8f __attribute__((ext_vector_type(8)));
typedef __attribute__((__vector_size__(4 * sizeof(int)))) int vsi4;

#define N_NODES 100000
#define N_EDGES 1600000
#define CH      128
#define ROW_TILES (N_NODES / 16)   // 6250, exact
#define PW_FLOATS (32 * 8 * 32 * 2) // 16384 floats = 64 KB packed W

#if defined(__has_builtin)
#if __has_builtin(__builtin_amdgcn_global_load_async_to_lds_b128) && \
    __has_builtin(__builtin_amdgcn_s_wait_asynccnt)
#define USE_ASYNC_LDS 1
#endif
#endif

// ---------------------------------------------------------------------------
// Prepack W[128x128] (row-major K x N) into per-lane WMMA B-fragment layout,
// two K-steps paired per 16B chunk:
//   pw[(((kp*8 + t)*32 + l)*4) + q2]
//   q2 = (ks&1)*2 + f,  ks = kp*2 + (q2>>1),  k = ks*4 + (l>>4)*2 + f,
//   n = t*16 + (l&15)
// ---------------------------------------------------------------------------
__global__ void pack_w(const float* __restrict__ W, float* __restrict__ pw) {
    int i = blockIdx.x * blockDim.x + threadIdx.x;
    if (i >= PW_FLOATS) return;
    int q2 = i & 3;
    int l  = (i >> 2) & 31;
    int p  = i >> 7;           // 0..127
    int kp = p >> 3;
    int t  = p & 7;
    int f  = q2 & 1;
    int ks = kp * 2 + (q2 >> 1);
    int k  = ks * 4 + (l >> 4) * 2 + f;
    int n  = t * 16 + (l & 15);
    pw[i] = W[k * CH + n];
}

// ---------------------------------------------------------------------------
// h = x @ W via V_WMMA_F32_16X16X4_F32. One wave -> 16 rows x 128 cols.
// Packed W staged to LDS once per block (async-to-LDS when available),
// B fragments read as conflict-free ds_load_b128 (one pair of K-steps).
// ---------------------------------------------------------------------------
__global__ __launch_bounds__(256) void gemm_wmma_f32(const float* __restrict__ x,
                                                     const float* __restrict__ pw,
                                                     float* __restrict__ h) {
    __shared__ float4 lds_w4[PW_FLOATS / 4];   // 64 KB

    // ---- cooperative stage: packed W -> LDS (all 8 waves participate) ----
#ifdef USE_ASYNC_LDS
    for (int i = threadIdx.x; i < PW_FLOATS / 4; i += 256) {
        __builtin_amdgcn_global_load_async_to_lds_b128(
            (__attribute__((address_space(1))) vsi4*)(pw + 4 * i),
            (__attribute__((address_space(3))) vsi4*)&lds_w4[i],
            0, 0);
    }
    __builtin_amdgcn_s_wait_asynccnt(0);
#else
    for (int i = threadIdx.x; i < PW_FLOATS / 4; i += 256) {
        lds_w4[i] = ((const float4*)pw)[i];
    }
#endif
    __syncthreads();

    const int tile = blockIdx.x * 8 + (threadIdx.x >> 5);
    if (tile >= ROW_TILES) return;             // wave-uniform (after the barrier)

    const int lane = threadIdx.x & 31;
    const int n    = lane & 15;                // A row / D col
    const int hk   = (lane >> 4) * 2;          // K sub-offset: 0 or 2
    const int row0 = tile * 16;
    const float* xrow = x + (size_t)(row0 + n) * CH;

    v8f acc[8] = {};

    for (int kp = 0; kp < 16; ++kp) {          // pair of K-steps: ks = 2kp, 2kp+1
        const int kb0 = kp * 8;                // K base of even step
        v2f a0, a1;
        a0.x = xrow[kb0 + hk];
        a0.y = xrow[kb0 + hk + 1];
        a1.x = xrow[kb0 + 4 + hk];
        a1.y = xrow[kb0 + 4 + hk + 1];
#pragma unroll
        for (int t = 0; t < 8; ++t) {
            v4f bb = *((const v4f*)&lds_w4[(kp * 8 + t) * 32 + lane]);
            v2f b0, b1;
            b0.x = bb.x; b0.y = bb.y;          // even K-step fragment
            b1.x = bb.z; b1.y = bb.w;          // odd  K-step fragment
            acc[t] = __builtin_amdgcn_wmma_f32_16x16x4_f32(
                false, a0, false, b0, (short)0, acc[t], false, false);
            acc[t] = __builtin_amdgcn_wmma_f32_16x16x4_f32(
                false, a1, false, b1, (short)0, acc[t], false, false);
        }
    }

    const int mbase = row0 + (lane >> 4) * 8;
#pragma unroll
    for (int t = 0; t < 8; ++t) {
#pragma unroll
        for (int v = 0; v < 8; ++v) {
            h[(size_t)(mbase + v) * CH + t * 16 + n] = acc[t][v];
        }
    }
}

// deg[i] = 1.0 (self-loop contribution)
__global__ void deg_init(float* __restrict__ deg) {
    int i = blockIdx.x * blockDim.x + threadIdx.x;
    if (i < N_NODES) deg[i] = 1.0f;
}

// deg[dst] += 1 for every edge (f32 atomic; exact up to 2^24 >> 1.6M)
__global__ void deg_count(const int* __restrict__ dst, float* __restrict__ deg) {
    int e = blockIdx.x * blockDim.x + threadIdx.x;
    if (e < N_EDGES) atomicAdd(&deg[dst[e]], 1.0f);
}

// deg -> d^{-1/2} in place
__global__ void deg_to_dinv(float* __restrict__ deg) {
    int i = blockIdx.x * blockDim.x + threadIdx.x;
    if (i < N_NODES) {
        float d = deg[i];
        deg[i] = (d > 0.0f) ? rsqrtf(d) : 0.0f;
    }
}

// out = h * dinv^2 (self-loop message) + bias (also initializes d_out)
__global__ void out_init(const float* __restrict__ h, const float* __restrict__ dinv,
                         const float* __restrict__ bias, float* __restrict__ out) {
    int g = blockIdx.x * blockDim.x + threadIdx.x;
    if (g >= N_NODES * 32) return;
    int node = g >> 5;
    int c4   = (g & 31) * 4;
    float w = dinv[node]; w *= w;
    const float4 hv = *(const float4*)(h + (size_t)node * CH + c4);
    const float4 bv = *(const float4*)(bias + c4);
    float4 o;
    o.x = hv.x * w + bv.x;
    o.y = hv.y * w + bv.y;
    o.z = hv.z * w + bv.z;
    o.w = hv.w * w + bv.w;
    *(float4*)(out + (size_t)node * CH + c4) = o;
}

// out[dst] += h[src] * dinv[src]*dinv[dst]; one wave per edge, float4 per lane.
// h and out are both L2-resident (102 MB total < 192 MB L2): atomics stay on-chip.
__global__ void edge_scatter(const int* __restrict__ ei, const float* __restrict__ h,
                             const float* __restrict__ dinv, float* __restrict__ out) {
    int g = blockIdx.x * blockDim.x + threadIdx.x;
    int e = g >> 5;
    if (e >= N_EDGES) return;
    int c4 = (g & 31) * 4;
    int s = ei[e];                // edge_index[0][e]  (source j)
    int d = ei[N_EDGES + e];      // edge_index[1][e]  (target i)
    float w = dinv[s] * dinv[d];
    const float4 hv = *(const float4*)(h + (size_t)s * CH + c4);
    float* op = out + (size_t)d * CH + c4;
    atomicAdd(op + 0, hv.x * w);
    atomicAdd(op + 1, hv.y * w);
    atomicAdd(op + 2, hv.z * w);
    atomicAdd(op + 3, hv.w * w);
}

// per-channel PReLU in place
__global__ void prelu_k(float* __restrict__ out, const float* __restrict__ slope) {
    int g = blockIdx.x * blockDim.x + threadIdx.x;
    if (g >= N_NODES * 32) return;
    int node = g >> 5;
    int c4   = (g & 31) * 4;
    float* p = out + (size_t)node * CH + c4;
    float4 v = *(float4*)p;
    const float4 av = *(const float4*)(slope + c4);
    v.x = v.x > 0.0f ? v.x : av.x * v.x;
    v.y = v.y > 0.0f ? v.y : av.y * v.y;
    v.z = v.z > 0.0f ? v.z : av.z * v.z;
    v.w = v.w > 0.0f ? v.w : av.w * v.w;
    *(float4*)p = v;
}

extern "C" void kernel_launch(void* const* d_in, const int* in_sizes, int n_in,
                              void* d_out, int out_size, void* d_ws, size_t ws_size,
                              hipStream_t stream) {
    const float* x    = (const float*)d_in[0];
    const int*   ei   = (const int*)  d_in[1];   // [2, E]: src row then dst row
    const float* W    = (const float*)d_in[2];
    const float* bias = (const float*)d_in[3];
    const float* a    = (const float*)d_in[4];
    float* out = (float*)d_out;

    // workspace layout: h [N*CH] | deg/dinv [N] | packed W [16384]
    float* h   = (float*)d_ws;
    float* deg = h + (size_t)N_NODES * CH;
    float* pw  = deg + N_NODES;

    pack_w      <<<(PW_FLOATS + 255) / 256, 256, 0, stream>>>(W, pw);
    gemm_wmma_f32<<<(ROW_TILES + 7) / 8, 256, 0, stream>>>(x, pw, h);
    deg_init    <<<(N_NODES + 255) / 256, 256, 0, stream>>>(deg);
    deg_count   <<<(N_EDGES + 255) / 256, 256, 0, stream>>>(ei + N_EDGES, deg);
    deg_to_dinv <<<(N_NODES + 255) / 256, 256, 0, stream>>>(deg);
    out_init    <<<(N_NODES * 32 + 255) / 256, 256, 0, stream>>>(h, deg, bias, out);
    edge_scatter<<<(N_EDGES * 32 + 255) / 256, 256, 0, stream>>>(ei, h, deg, out);
    prelu_k     <<<(N_NODES * 32 + 255) / 256, 256, 0, stream>>>(out, a);
}